// CollapsePool_58033598104035
// MI455X (gfx1250) — compile-verified
//
#include <hip/hip_runtime.h>
#include <hip/hip_bf16.h>

typedef float v2f __attribute__((ext_vector_type(2)));
typedef float v8f __attribute__((ext_vector_type(8)));

#define FDIM 256

// ---------------------------------------------------------------------------
// Phase 1: mag0[i] = sum_k x[i,k]^2 via V_WMMA_F32_16X16X4_F32 (full f32).
// Each wave handles a 16-row tile; rows staged through LDS (coalesced global
// loads), A = squared values, B = all-ones -> D[m][n] = row sum for every n.
// ---------------------------------------------------------------------------
__global__ __launch_bounds__(64) void mag_wmma_kernel(
    const float* __restrict__ x, float* __restrict__ mag, int N) {
  __shared__ float lds[2 * 16 * FDIM];  // 2 waves * 16 rows * 256 f32 = 32KB
  const int wave = threadIdx.x >> 5;
  const int lane = threadIdx.x & 31;
  const int tile = blockIdx.x * 2 + wave;
  const int base = tile * 16;
  const bool valid = (base + 16) <= N;   // uniform per wave
  float* tl = &lds[wave * 16 * FDIM];

  if (valid) {
    const float4* g = reinterpret_cast<const float4*>(x + (size_t)base * FDIM);
    float4* l4 = reinterpret_cast<float4*>(tl);
    for (int i = lane; i < 16 * FDIM / 4; i += 32) l4[i] = g[i];
  }
  __syncthreads();

  if (valid) {
    v8f acc = {0.f, 0.f, 0.f, 0.f, 0.f, 0.f, 0.f, 0.f};
    v2f bones; bones.x = 1.0f; bones.y = 1.0f;       // B = ones (layout-free)
    const int row = lane & 15;
    const int khalf = (lane >> 4) * 2;               // lanes 0-15: K0,K1; 16-31: K2,K3
    for (int k0 = 0; k0 < FDIM; k0 += 4) {
      float a0 = tl[row * FDIM + k0 + khalf];
      float a1 = tl[row * FDIM + k0 + khalf + 1];
      v2f a; a.x = a0 * a0; a.y = a1 * a1;
      acc = __builtin_amdgcn_wmma_f32_16x16x4_f32(
          false, a, false, bones, (short)0, acc, false, false);
    }
    // C/D layout: lanes 0-15 VGPR r -> M=r (col n=lane); lanes 16-31 -> M=r+8.
    if (lane == 0)  for (int r = 0; r < 8; ++r) mag[base + r]     = acc[r];
    if (lane == 16) for (int r = 0; r < 8; ++r) mag[base + 8 + r] = acc[r];
  }
}

// ---------------------------------------------------------------------------
// Init working state (re-run on every call -> deterministic replays).
// ---------------------------------------------------------------------------
__global__ void init_kernel(const int* __restrict__ edge_index,
                            int* srcw, int* dstw, unsigned char* ealive,
                            int* nbr_mark, int* node_alive, int* mod,
                            unsigned long long* best, int E, int N, int S) {
  int i = blockIdx.x * blockDim.x + threadIdx.x;
  if (i < E) { srcw[i] = edge_index[i]; dstw[i] = edge_index[E + i]; ealive[i] = 1; }
  if (i < N) { nbr_mark[i] = -1; node_alive[i] = 1; mod[i] = -1; }
  if (i < S) best[i] = ~0ull;
}

// ---------------------------------------------------------------------------
// Per-graph: K smallest mag0 ascending (stable -> key packs node id).
// ---------------------------------------------------------------------------
__global__ __launch_bounds__(1024) void select_kernel(
    const float* __restrict__ mag, const int* __restrict__ numg_p,
    int N, int S, int* __restrict__ schedule) {
  __shared__ unsigned long long red[1024];
  __shared__ unsigned long long prevs;
  const int g = blockIdx.x;
  const int numg = *numg_p;
  if (g >= numg) return;
  const int n_per = N / numg;
  const int K = S / numg;
  const int t = threadIdx.x;
  const int start = g * n_per;
  if (t == 0) prevs = 0ull;
  __syncthreads();
  for (int k = 0; k < K; ++k) {
    unsigned long long prev = prevs;
    unsigned long long lmin = ~0ull;
    for (int v = start + t; v < start + n_per; v += 1024) {
      unsigned long long key =
          ((unsigned long long)__float_as_uint(mag[v]) << 32) | (unsigned)v;
      if (key > prev && key < lmin) lmin = key;
    }
    red[t] = lmin;
    __syncthreads();
    for (int off = 512; off > 0; off >>= 1) {
      if (t < off && red[t + off] < red[t]) red[t] = red[t + off];
      __syncthreads();
    }
    if (t == 0) { prevs = red[0]; schedule[g * K + k] = (int)(red[0] & 0xffffffffu); }
    __syncthreads();
  }
}

__global__ void markdead_kernel(const int* __restrict__ schedule,
                                int* node_alive, int S) {
  int i = blockIdx.x * blockDim.x + threadIdx.x;
  if (i < S) node_alive[schedule[i]] = 0;
}

// ---------------------------------------------------------------------------
// Step kernels (3 launches per collapse step, stream-ordered).
// ---------------------------------------------------------------------------
__global__ void k1_find(const int* __restrict__ srcw, const int* __restrict__ dstw,
                        const unsigned char* __restrict__ ealive,
                        const float* __restrict__ mag, const int* __restrict__ schedule,
                        unsigned long long* best, int E, int s) {
  int e = blockIdx.x * blockDim.x + threadIdx.x;
  if (e >= E || !ealive[e]) return;
  int old = schedule[s];
  if (srcw[e] != old) return;
  int d = dstw[e];
  if (d == old) return;
  unsigned long long key =
      ((unsigned long long)__float_as_uint(mag[d]) << 32) | (unsigned)e;
  atomicMin(&best[s], key);
}

__global__ void k2_mark_update(const int* __restrict__ srcw, const int* __restrict__ dstw,
                               const unsigned char* __restrict__ ealive,
                               int* nbr_mark, const int* __restrict__ schedule,
                               const unsigned long long* __restrict__ best,
                               const float* __restrict__ x, const float* __restrict__ pos,
                               float* xov, float* posov, int* mod, int* newv,
                               int E, int s) {
  __shared__ int s_newslot, s_oldslot, s_fresh;
  unsigned long long b = best[s];
  if (b == ~0ull) return;
  const int emin = (int)(b & 0xffffffffu);
  const int nw = dstw[emin];            // stable: nothing rewrites dstw here
  if ((int)blockIdx.x < (int)gridDim.x - 1) {
    // mark neighbors of `new` BEFORE any rewiring (matches stale nbr_of_new)
    int e = blockIdx.x * blockDim.x + threadIdx.x;
    if (e < E && ealive[e] && srcw[e] == nw) nbr_mark[dstw[e]] = s;
  } else {
    const int old = schedule[s];
    if (threadIdx.x == 0) {
      newv[s] = nw;
      int m = mod[nw];
      if (m < 0) { mod[nw] = s; m = s; s_fresh = 1; } else { s_fresh = 0; }
      s_newslot = m;
      s_oldslot = mod[old];
    }
    __syncthreads();
    int f = threadIdx.x;
    if (f < FDIM) {
      float ov = (s_oldslot >= 0) ? xov[(size_t)s_oldslot * FDIM + f]
                                  : x[(size_t)old * FDIM + f];
      float nv = s_fresh ? x[(size_t)nw * FDIM + f]
                         : xov[(size_t)s_newslot * FDIM + f];
      xov[(size_t)s_newslot * FDIM + f] = nv + 0.5f * ov;
    }
    if (f < 3) {
      float ov = (s_oldslot >= 0) ? posov[s_oldslot * 3 + f] : pos[(size_t)old * 3 + f];
      float nv = s_fresh ? pos[(size_t)nw * 3 + f] : posov[s_newslot * 3 + f];
      posov[s_newslot * 3 + f] = nv + 0.5f * ov;
    }
  }
}

__global__ void k3_rewire(int* srcw, int* dstw, unsigned char* ealive,
                          const int* __restrict__ nbr_mark,
                          const int* __restrict__ schedule,
                          const unsigned long long* __restrict__ best,
                          const int* __restrict__ newv, int E, int s) {
  int e = blockIdx.x * blockDim.x + threadIdx.x;
  if (e >= E || !ealive[e]) return;
  if (best[s] == ~0ull) return;
  const int old = schedule[s];
  const int nw = newv[s];
  int se = srcw[e], de = dstw[e];
  if (se == old) {            // loop 1: edges (old, d)
    if (de == nw || de == old || nbr_mark[de] == s) ealive[e] = 0;
    else srcw[e] = nw;
  } else if (de == old) {     // loop 2: edges (s, old)  (disjoint: no self-loops)
    if (se == nw || se == old || nbr_mark[se] == s) ealive[e] = 0;
    else dstw[e] = nw;
  }
}

// ---------------------------------------------------------------------------
// Compaction: block sums -> exclusive scan -> remap + pos/batch scatter.
// ---------------------------------------------------------------------------
__global__ void blocksum_kernel(const int* __restrict__ node_alive, int* bsum, int N) {
  __shared__ int sh[256];
  int v = blockIdx.x * 256 + threadIdx.x;
  sh[threadIdx.x] = (v < N) ? node_alive[v] : 0;
  __syncthreads();
  for (int off = 128; off > 0; off >>= 1) {
    if ((int)threadIdx.x < off) sh[threadIdx.x] += sh[threadIdx.x + off];
    __syncthreads();
  }
  if (threadIdx.x == 0) bsum[blockIdx.x] = sh[0];
}

__global__ void scan_bsum_kernel(int* bsum, int nb) {
  if (blockIdx.x == 0 && threadIdx.x == 0) {
    int run = 0;
    for (int i = 0; i < nb; ++i) { int t = bsum[i]; bsum[i] = run; run += t; }
  }
}

__global__ void remap_kernel(const int* __restrict__ node_alive,
                             const int* __restrict__ bsum, int* remapw,
                             const float* __restrict__ pos,
                             const float* __restrict__ posov,
                             const int* __restrict__ mod,
                             const int* __restrict__ batch,
                             float* pos_out, float* batch_out, int N) {
  __shared__ int sh[256];
  const int t = threadIdx.x;
  const int v = blockIdx.x * 256 + t;
  const int a = (v < N) ? node_alive[v] : 0;
  sh[t] = a;
  __syncthreads();
  for (int off = 1; off < 256; off <<= 1) {   // inclusive Hillis-Steele
    int add = (t >= off) ? sh[t - off] : 0;
    __syncthreads();
    sh[t] += add;
    __syncthreads();
  }
  if (v < N && a) {
    int p = bsum[blockIdx.x] + sh[t] - 1;     // exclusive position
    remapw[v] = p;
    int m = mod[v];
    const float* pr = (m >= 0) ? (posov + (size_t)m * 3) : (pos + (size_t)v * 3);
    pos_out[p * 3 + 0] = pr[0];
    pos_out[p * 3 + 1] = pr[1];
    pos_out[p * 3 + 2] = pr[2];
    batch_out[p] = (float)batch[v];
  }
}

__global__ __launch_bounds__(64) void xgather_kernel(
    const float* __restrict__ x, const float* __restrict__ xov,
    const int* __restrict__ mod, const int* __restrict__ node_alive,
    const int* __restrict__ remapw, float* __restrict__ x_out, int N) {
  const int v = blockIdx.x;
  if (!node_alive[v]) return;
  const int p = remapw[v];
  const int m = mod[v];
  const float4* src = (m >= 0)
      ? reinterpret_cast<const float4*>(xov + (size_t)m * FDIM)
      : reinterpret_cast<const float4*>(x + (size_t)v * FDIM);
  float4* dst = reinterpret_cast<float4*>(x_out + (size_t)p * FDIM);
  int vn = v + 64; if (vn >= N) vn = N - 1;
  __builtin_prefetch(x + (size_t)vn * FDIM, 0, 1);   // global_prefetch_b8
  for (int i = threadIdx.x; i < FDIM / 4; i += 64) dst[i] = src[i];
}

__global__ void eout_kernel(const int* __restrict__ srcw, const int* __restrict__ dstw,
                            const unsigned char* __restrict__ ealive,
                            const int* __restrict__ remapw, float* e_out, int E) {
  int e = blockIdx.x * blockDim.x + threadIdx.x;
  if (e >= E) return;
  if (ealive[e]) {
    e_out[e]     = (float)remapw[srcw[e]];
    e_out[E + e] = (float)remapw[dstw[e]];
  } else {
    e_out[e] = -1.0f;
    e_out[E + e] = -1.0f;
  }
}

// ---------------------------------------------------------------------------
extern "C" void kernel_launch(void* const* d_in, const int* in_sizes, int n_in,
                              void* d_out, int out_size, void* d_ws, size_t ws_size,
                              hipStream_t stream) {
  const float* x          = (const float*)d_in[0];
  const float* pos        = (const float*)d_in[1];
  const int*   edge_index = (const int*)d_in[2];
  const int*   batch      = (const int*)d_in[4];
  const int*   numg_p     = (const int*)d_in[5];

  const int N = in_sizes[4];
  const int F = in_sizes[0] / N;          // == FDIM (256) for this problem
  const int E = in_sizes[2] / 2;
  const int n_out = (out_size - 2 * E) / (F + 4);
  const int S = N - n_out;                // total sequential collapse steps
  (void)F; (void)ws_size;

  // workspace carve-out (aligned to 256B)
  char* w = (char*)d_ws;
  auto carve = [&](size_t bytes) {
    void* p = (void*)w;
    w += (bytes + 255) & ~(size_t)255;
    return p;
  };
  float*              mag        = (float*)carve((size_t)N * 4);
  int*                schedule   = (int*)carve((size_t)S * 4);
  unsigned long long* best       = (unsigned long long*)carve((size_t)S * 8);
  int*                newv       = (int*)carve((size_t)S * 4);
  int*                srcw       = (int*)carve((size_t)E * 4);
  int*                dstw       = (int*)carve((size_t)E * 4);
  unsigned char*      ealive     = (unsigned char*)carve((size_t)E);
  int*                nbr_mark   = (int*)carve((size_t)N * 4);
  int*                node_alive = (int*)carve((size_t)N * 4);
  int*                mod        = (int*)carve((size_t)N * 4);
  int*                remapw     = (int*)carve((size_t)N * 4);
  int*                bsum       = (int*)carve((size_t)((N + 255) / 256) * 4);
  float*              xov        = (float*)carve((size_t)S * FDIM * 4);
  float*              posov      = (float*)carve((size_t)S * 3 * 4);

  // output layout (flat, return order): x2 | pos2 | e_out | batch
  float* out       = (float*)d_out;
  float* x_out     = out;
  float* pos_out   = out + (size_t)n_out * FDIM;
  float* e_out     = pos_out + (size_t)n_out * 3;
  float* batch_out = e_out + (size_t)2 * E;

  int mx = E > N ? E : N; if (S > mx) mx = S;
  init_kernel<<<(mx + 255) / 256, 256, 0, stream>>>(
      edge_index, srcw, dstw, ealive, nbr_mark, node_alive, mod, best, E, N, S);

  mag_wmma_kernel<<<(N + 31) / 32, 64, 0, stream>>>(x, mag, N);

  select_kernel<<<32, 1024, 0, stream>>>(mag, numg_p, N, S, schedule);
  markdead_kernel<<<(S + 255) / 256, 256, 0, stream>>>(schedule, node_alive, S);

  const int nEB = (E + 255) / 256;
  for (int s = 0; s < S; ++s) {
    k1_find<<<nEB, 256, 0, stream>>>(srcw, dstw, ealive, mag, schedule, best, E, s);
    k2_mark_update<<<nEB + 1, 256, 0, stream>>>(srcw, dstw, ealive, nbr_mark, schedule,
                                                best, x, pos, xov, posov, mod, newv, E, s);
    k3_rewire<<<nEB, 256, 0, stream>>>(srcw, dstw, ealive, nbr_mark, schedule, best,
                                       newv, E, s);
  }

  const int nb = (N + 255) / 256;
  blocksum_kernel<<<nb, 256, 0, stream>>>(node_alive, bsum, N);
  scan_bsum_kernel<<<1, 32, 0, stream>>>(bsum, nb);
  remap_kernel<<<nb, 256, 0, stream>>>(node_alive, bsum, remapw, pos, posov, mod,
                                       batch, pos_out, batch_out, N);
  xgather_kernel<<<N, 64, 0, stream>>>(x, xov, mod, node_alive, remapw, x_out, N);
  eout_kernel<<<(E + 255) / 256, 256, 0, stream>>>(srcw, dstw, ealive, remapw, e_out, E);
}